// HSA_42545946034969
// MI455X (gfx1250) — compile-verified
//
#include <hip/hip_runtime.h>

#define FLEN 32
#define TCOLS 2048
#define NSTEPS (TCOLS - FLEN + 1)   // 2017
#define ROWS_PER_BLOCK 32
#define PAD 33                      // 33 coprime with 64 banks -> conflict-free
#define NCHUNK (TCOLS / 32)         // 64

__global__ __launch_bounds__(32) void hsa_spike_kernel(const float* __restrict__ sig,
                                                       const float* __restrict__ filt,
                                                       float* __restrict__ out) {
    // Double-buffered input staging + spike staging tile.
    __shared__ float inbuf[2][ROWS_PER_BLOCK * PAD];
    __shared__ float spbuf[ROWS_PER_BLOCK * PAD];

    const int lane = threadIdx.x;                     // one row per lane
    const int row0 = blockIdx.x * ROWS_PER_BLOCK;

    // f = 0.2 * filt (and negated copy so the masked subtract is a single FMA).
    float fv[FLEN], nfv[FLEN];
#pragma unroll
    for (int j = 0; j < FLEN; ++j) {
        float f = filt[j] * 0.2f;
        fv[j]  = f;
        nfv[j] = -f;
    }

    // Async-stage chunk k (columns 32k..32k+31 of rows row0..row0+31) into inbuf[k&1].
    // One coalesced 128B async load per row; LDS dest is row-major with PAD.
    auto prefetch_chunk = [&](int k) {
        const unsigned buf  = (unsigned)k & 1u;
        const unsigned voff = (unsigned)lane * 4u;    // byte offset along the row
#pragma unroll
        for (int r = 0; r < ROWS_PER_BLOCK; ++r) {
            const float* srcp = sig + (size_t)(row0 + r) * TCOLS + (size_t)k * 32;
            unsigned ldsa = (unsigned)(unsigned long long)&inbuf[buf][r * PAD + lane];
            asm volatile("global_load_async_to_lds_b32 %0, %1, %2"
                         :: "v"(ldsa), "v"(voff), "s"(srcp)
                         : "memory");
        }
    };
    auto wait_async = []() {
        asm volatile("s_wait_asynccnt 0" ::: "memory");
    };

    // Flush one 32x32 spike tile: conflict-free LDS reads, coalesced 128B stores.
    auto flush_out = [&](int t) {
#pragma unroll
        for (int r = 0; r < ROWS_PER_BLOCK; ++r) {
            out[(size_t)(row0 + r) * TCOLS + t * 32 + lane] = spbuf[r * PAD + lane];
        }
    };

    // Stage chunk 0, fill the register window, start chunk 1.
    prefetch_chunk(0);
    wait_async();
    float w[FLEN];                                    // rotating window registers
#pragma unroll
    for (int j = 0; j < FLEN; ++j) w[j] = inbuf[0][lane * PAD + j];
    prefetch_chunk(1);

    // Tiles 0..62: steps i = 32t + c, c = 0..31 (all valid, i <= 2015).
    for (int t = 0; t < NCHUNK - 1; ++t) {
        wait_async();                                 // chunk t+1 is now in LDS
        if (t < NCHUNK - 2) prefetch_chunk(t + 2);    // overlap next stage with compute
        const float* nb = &inbuf[(t + 1) & 1][0];

#pragma unroll
        for (int c = 0; c < 32; ++c) {
            // Invariant: w[(i+j)&31] == sig-residual column i+j, i = 32t + c.
            bool ok = true;
#pragma unroll
            for (int j = 0; j < FLEN; ++j)
                ok = ok & (w[(c + j) & 31] >= fv[j]);   // v_cmp + SALU mask AND-tree
            float m = ok ? 1.0f : 0.0f;
#pragma unroll
            for (int j = 0; j < FLEN; ++j)              // masked subtract: exact for m in {0,1}
                w[(c + j) & 31] = __builtin_fmaf(nfv[j], m, w[(c + j) & 31]);
            spbuf[lane * PAD + c] = m;                  // stage spike
            w[c] = nb[lane * PAD + c];                  // slide: bring in column i+32
        }
        flush_out(t);
    }

    // Tail tile 63: only i = 2016 is a real step; columns 2017..2047 are zeros.
    {
        bool ok = true;
#pragma unroll
        for (int j = 0; j < FLEN; ++j)
            ok = ok & (w[j] >= fv[j]);                  // window = columns 2016..2047
        spbuf[lane * PAD + 0] = ok ? 1.0f : 0.0f;       // no residual update needed (last step)
#pragma unroll
        for (int c = 1; c < 32; ++c) spbuf[lane * PAD + c] = 0.0f;
        flush_out(NCHUNK - 1);
    }
}

extern "C" void kernel_launch(void* const* d_in, const int* in_sizes, int n_in,
                              void* d_out, int out_size, void* d_ws, size_t ws_size,
                              hipStream_t stream) {
    const float* sig  = (const float*)d_in[0];   // [8192, 2048] f32
    const float* filt = (const float*)d_in[1];   // [32] f32
    float* out = (float*)d_out;                  // [8192, 2048] f32

    const int B = in_sizes[0] / TCOLS;           // 8192 rows
    dim3 grid(B / ROWS_PER_BLOCK);               // 256 single-wave blocks
    dim3 block(32);
    hipLaunchKernelGGL(hsa_spike_kernel, grid, block, 0, stream, sig, filt, out);
}